// SpGQAFlashDecodeAttention_18657337934297
// MI455X (gfx1250) — compile-verified
//
#include <hip/hip_runtime.h>
#include <cstdint>
#include <cstddef>

// ---------------- problem constants (from reference) ----------------
#define R_    8
#define B_    8
#define L_    2048
#define P_    16384
#define HQ    32
#define HKV   8
#define G_    4
#define DK    128
#define DV    128
#define SCALE 0.08838834764831845f
#define NEGBIG (-1e30f)

// split-K decode: each single-wave workgroup handles L_/SPLIT keys
#define SPLIT       4
#define KEYS_PER_WG (L_ / SPLIT)   // 512
#define NPART       (R_ * SPLIT)   // 32 partials per (b,h)

// ---------------- LDS layout (dynamic shared, offset 0 based) -------
// K/V tiles: 16 token rows x 128 floats, row padded to 132 dwords (528B)
// -> 16-lane fragment reads hit 16 distinct banks.
#define KROWB   528
#define KTILEB  (16 * KROWB)                 // 8448 B
#define KOFF(buf) ((unsigned)((buf) * KTILEB))
#define VOFF(buf) ((unsigned)(2 * KTILEB + (buf) * KTILEB))
#define PROW    18                            // P tile row stride (dwords)
#define POFF    ((unsigned)(4 * KTILEB))      // 33792
#define LDS_BYTES (4 * KTILEB + 16 * PROW * 4) // 34944

typedef __attribute__((ext_vector_type(2))) float v2f;
typedef __attribute__((ext_vector_type(8))) float v8f;

// Only check on the DEVICE pass: __has_builtin() for amdgcn builtins is
// false during the host (x86) pass even though the builtin works in device code.
#if defined(__HIP_DEVICE_COMPILE__)
#if !__has_builtin(__builtin_amdgcn_wmma_f32_16x16x4_f32)
#error "device pass: missing __builtin_amdgcn_wmma_f32_16x16x4_f32"
#endif
#endif

// one async 128B-per-8-lane DMA; same immediate offset is applied to both the
// global address and the LDS destination address (ISA 15.18.3).
#define GLDS(ldso, ga, OFFSTR)                                             \
  asm volatile("global_load_async_to_lds_b128 %0, %1, off offset:" OFFSTR  \
               :: "v"(ldso), "v"(ga) : "memory")

// copy one 16-row x 512B tile: each lane owns row (lane&15), 16B segment
// base (lane>>4)*16; 16 instructions stride 32B cover the full 512B row.
__device__ __forceinline__ void async_tile(unsigned ldso, uint64_t ga) {
  GLDS(ldso, ga, "0");   GLDS(ldso, ga, "32");  GLDS(ldso, ga, "64");
  GLDS(ldso, ga, "96");  GLDS(ldso, ga, "128"); GLDS(ldso, ga, "160");
  GLDS(ldso, ga, "192"); GLDS(ldso, ga, "224"); GLDS(ldso, ga, "256");
  GLDS(ldso, ga, "288"); GLDS(ldso, ga, "320"); GLDS(ldso, ga, "352");
  GLDS(ldso, ga, "384"); GLDS(ldso, ga, "416"); GLDS(ldso, ga, "448");
  GLDS(ldso, ga, "480");
}

__device__ __forceinline__ void wait_async0()  { asm volatile("s_wait_asynccnt 0x0"  ::: "memory"); }
__device__ __forceinline__ void wait_async32() { asm volatile("s_wait_asynccnt 0x20" ::: "memory"); }

// ------------------------------------------------------------------
// Kernel 1: per-(r, b, h, split) flash-decode partial.
// One wave (32 lanes). Writes unnormalized O (16x128, rows 0..3 valid),
// running row-max m and row-sum l to workspace.
// ------------------------------------------------------------------
__global__ __launch_bounds__(32)
void SpGQA_partial(const float* __restrict__ q,
                   const float* __restrict__ kc,
                   const float* __restrict__ vc,
                   const int*   __restrict__ gkv,
                   const int*   __restrict__ bt,
                   float*       __restrict__ ws) {
  extern __shared__ char smem[];

  const int lane = threadIdx.x;
  const int half = lane >> 4;      // 0: lanes 0-15, 1: lanes 16-31
  const int l16  = lane & 15;

  const int wg = blockIdx.x;
  const int s  = wg % SPLIT;
  const int h  = (wg / SPLIT) % HKV;
  const int b  = (wg / (SPLIT * HKV)) % B_;
  const int r  = wg / (SPLIT * HKV * B_);
  const int part = r * SPLIT + s;

  const int kv_len   = gkv[r * B_ + b];
  const int key_base = s * KEYS_PER_WG;

  float* Ows = ws;                                                // [B*HKV*NPART][G*DV]
  float* Mws = ws + (size_t)B_ * HKV * NPART * G_ * DV;           // [B*HKV*NPART][G]
  float* Lws = Mws + (size_t)B_ * HKV * NPART * G_;
  const size_t pidx = (size_t)(b * HKV + h) * NPART + part;

  if (key_base >= kv_len) {                       // wave-uniform: dead split
    if (lane < 16) {
      #pragma unroll
      for (int i = 0; i < G_; ++i)
        #pragma unroll
        for (int t = 0; t < 8; ++t)
          Ows[pidx * G_ * DV + i * DV + t * 16 + l16] = 0.f;
    }
    if (lane == 0) {
      #pragma unroll
      for (int i = 0; i < G_; ++i) { Mws[pidx * G_ + i] = NEGBIG; Lws[pidx * G_ + i] = 0.f; }
    }
    return;
  }

  // ---- preload Q A-fragments (M=16 rows: rows 0..3 real, rest zero) ----
  // A layout: lanes 0-15 row=l16 cols {4kk, 4kk+1}; lanes 16-31 cols {4kk+2, 4kk+3}
  v2f qa[32];
  {
    const int row = l16;
    const bool qv = row < G_;
    const float* qrow = q + ((size_t)b * HQ + h * G_ + (qv ? row : 0)) * DK;
    #pragma unroll
    for (int kk = 0; kk < 32; ++kk) {
      const int d0 = kk * 4 + (half ? 2 : 0);
      v2f t = *(const v2f*)(qrow + d0);
      qa[kk] = qv ? t : (v2f){0.f, 0.f};
    }
  }

  // ---- online softmax state (per lane: rows i (+8 for half 1)) ----
  float mrow[8], lrow[8];
  v8f   Oc[8];
  const v8f vzero = {0.f, 0.f, 0.f, 0.f, 0.f, 0.f, 0.f, 0.f};
  #pragma unroll
  for (int i = 0; i < 8; ++i) { mrow[i] = NEGBIG; lrow[i] = 0.f; }
  #pragma unroll
  for (int t = 0; t < 8; ++t) Oc[t] = vzero;

  const int*   btrow  = bt + ((size_t)r * B_ + b) * L_ + key_base;
  const uint64_t kbase = (uint64_t)(uintptr_t)(kc + (size_t)r * P_ * HKV * DK);
  const uint64_t vbase = (uint64_t)(uintptr_t)(vc + (size_t)r * P_ * HKV * DV);

  auto issue = [&](int c, int buf) {
    const int tok = btrow[c * 16 + l16];                 // lanes 16-31 mirror 0-15
    const uint64_t roff = ((uint64_t)tok * HKV + h) * (DK * 4) + (uint64_t)(half << 4);
    async_tile(KOFF(buf) + (unsigned)(l16 * KROWB + (half << 4)), kbase + roff);
    async_tile(VOFF(buf) + (unsigned)(l16 * KROWB + (half << 4)), vbase + roff);
  };

  auto compute = [&](int c, int buf) {
    // ---- QK: S(16x16) = Q(16x128) x K^T(128x16), 32 wmma k-steps ----
    v8f S = vzero;
    const char* kb = smem + KOFF(buf);
    #pragma unroll
    for (int kk = 0; kk < 32; ++kk) {
      const int d0 = kk * 4 + (half ? 2 : 0);
      // B layout mirror of A: lanes = N (key), VGPR pair = K rows {d0, d0+1}
      v2f bf = *(const v2f*)(kb + l16 * KROWB + d0 * 4);
      S = __builtin_amdgcn_wmma_f32_16x16x4_f32(false, qa[kk], false, bf,
                                                (short)0, S, false, false);
    }

    const int  keyidx = key_base + c * 16 + l16;         // column of this lane
    const bool vcol   = keyidx < kv_len;

    // ---- online softmax per row; emit P tile to LDS ----
    #pragma unroll
    for (int i = 0; i < 8; ++i) {
      float sc = vcol ? S[i] * SCALE : NEGBIG;
      float mc = sc;
      #pragma unroll
      for (int w = 1; w < 16; w <<= 1) mc = fmaxf(mc, __shfl_xor(mc, w, 32));
      const float mnew  = fmaxf(mrow[i], mc);
      const float alpha = __expf(mrow[i] - mnew);
      const float p     = vcol ? __expf(sc - mnew) : 0.f;
      float ps = p;
      #pragma unroll
      for (int w = 1; w < 16; w <<= 1) ps += __shfl_xor(ps, w, 32);
      lrow[i] = lrow[i] * alpha + ps;
      mrow[i] = mnew;
      #pragma unroll
      for (int t = 0; t < 8; ++t) Oc[t][i] *= alpha;
      const int prow = i + half * 8;
      *(float*)(smem + POFF + (unsigned)((prow * PROW + l16) * 4)) = p;
    }

    // ---- PV: O(16x128) += P(16x16) x V(16x128), 4 k-steps x 8 n-tiles ----
    const char* vb = smem + VOFF(buf);
    #pragma unroll
    for (int j = 0; j < 4; ++j) {
      const int c0 = j * 4 + (half ? 2 : 0);
      v2f pa = *(const v2f*)(smem + POFF + (unsigned)((l16 * PROW + c0) * 4));
      #pragma unroll
      for (int t = 0; t < 8; ++t) {
        const int col = t * 16 + l16;
        v2f bf;
        bf.x = *(const float*)(vb + (c0    ) * KROWB + col * 4);
        bf.y = *(const float*)(vb + (c0 + 1) * KROWB + col * 4);
        Oc[t] = __builtin_amdgcn_wmma_f32_16x16x4_f32(false, pa, false, bf,
                                                      (short)0, Oc[t], false, false);
      }
    }
  };

  const int remaining = min(kv_len - key_base, KEYS_PER_WG);
  const int nact = (remaining + 15) >> 4;

  issue(0, 0);
  for (int c = 0; c < nact; ++c) {
    if (c + 1 < nact) { issue(c + 1, (c + 1) & 1); wait_async32(); }
    else              { wait_async0(); }
    compute(c, c & 1);
  }

  // ---- store partial: rows 0..3 live in VGPR i (lanes 0-15) ----
  if (lane < 16) {
    #pragma unroll
    for (int i = 0; i < G_; ++i)
      #pragma unroll
      for (int t = 0; t < 8; ++t)
        Ows[pidx * G_ * DV + i * DV + t * 16 + l16] = Oc[t][i];
  }
  if (lane == 0) {
    #pragma unroll
    for (int i = 0; i < G_; ++i) { Mws[pidx * G_ + i] = mrow[i]; Lws[pidx * G_ + i] = lrow[i]; }
  }
}

// ------------------------------------------------------------------
// Kernel 2: LSE-merge the NPART partials per (b, h, g) -> final output.
// Exactly the reference math: o_p = O_p/max(l,eps), lse_p = m + log(max(l,eps)).
// ------------------------------------------------------------------
__global__ __launch_bounds__(DV)
void SpGQA_combine(const float* __restrict__ ws, float* __restrict__ out) {
  const int g = blockIdx.x & (G_ - 1);
  const int h = (blockIdx.x >> 2) & (HKV - 1);
  const int b = blockIdx.x >> 5;
  const int d = threadIdx.x;

  const float* Ows = ws;
  const float* Mws = ws + (size_t)B_ * HKV * NPART * G_ * DV;
  const float* Lws = Mws + (size_t)B_ * HKV * NPART * G_;
  const size_t base = (size_t)(b * HKV + h) * NPART;

  float gm = NEGBIG;
  for (int p = 0; p < NPART; ++p) {
    const float m = Mws[(base + p) * G_ + g];
    const float l = Lws[(base + p) * G_ + g];
    gm = fmaxf(gm, m + __logf(fmaxf(l, 1e-30f)));
  }
  float acc = 0.f, wsum = 0.f;
  for (int p = 0; p < NPART; ++p) {
    const float m = Mws[(base + p) * G_ + g];
    const float l = Lws[(base + p) * G_ + g];
    const float w = __expf(m + __logf(fmaxf(l, 1e-30f)) - gm);
    const float o = Ows[((base + p) * G_ + g) * DV + d] / fmaxf(l, 1e-30f);
    acc += w * o;
    wsum += w;
  }
  out[((size_t)b * HQ + h * G_ + g) * DV + d] = acc / fmaxf(wsum, 1e-30f);
}

extern "C" void kernel_launch(void* const* d_in, const int* in_sizes, int n_in,
                              void* d_out, int out_size, void* d_ws, size_t ws_size,
                              hipStream_t stream) {
  (void)in_sizes; (void)n_in; (void)out_size; (void)ws_size;
  const float* q   = (const float*)d_in[0];
  const float* kc  = (const float*)d_in[1];
  const float* vc  = (const float*)d_in[2];
  const int*   gkv = (const int*)d_in[3];
  const int*   bt  = (const int*)d_in[4];
  float* out = (float*)d_out;
  float* ws  = (float*)d_ws;

  SpGQA_partial<<<dim3(R_ * B_ * HKV * SPLIT), dim3(32), LDS_BYTES, stream>>>(
      q, kc, vc, gkv, bt, ws);
  SpGQA_combine<<<dim3(B_ * HKV * G_), dim3(DV), 0, stream>>>(ws, out);
}